// BayesianLinearLayer_58695023067356
// MI455X (gfx1250) — compile-verified
//
#include <hip/hip_runtime.h>
#include <hip/hip_bf16.h>

typedef __attribute__((ext_vector_type(2))) float v2f;
typedef __attribute__((ext_vector_type(4))) float v4f;
typedef __attribute__((ext_vector_type(8))) float v8f;

#define LOG_SQRT_2PI 0.9189385332046727f
// log(0.5) - LOG_SQRT_2PI
#define A1C (-1.6120857137646181f)
// delta = log((1-pi)/pi * sigma1/sigma2) - (1/(2*s2^2) - 1/(2*s1^2)) * w^2
//       = log(500) - 124999.5 * w^2
#define DELTA0 6.2146080984221914f
#define DELTAK (-124999.5f)

__device__ __forceinline__ float softplusf(float x) {
    return fmaxf(x, 0.f) + log1pf(__expf(-fabsf(x)));
}

// log( 0.5*N(w;0,1) + 0.5*N(w;0,0.002) ) via stable logaddexp
__device__ __forceinline__ float mix_lp(float w) {
    float v2 = w * w;
    float delta = fmaf(DELTAK, v2, DELTA0);
    return A1C - 0.5f * v2 + fmaxf(delta, 0.f) + log1pf(__expf(-fabsf(delta)));
}

// ---------------------------------------------------------------------------
// Kernel 1: sigma_w = softplus(weight_rho); per-block partial of sum(log sigma)
// 1M elements -> 4096 blocks x 256 threads
// ---------------------------------------------------------------------------
__global__ __launch_bounds__(256) void prep_kernel(const float* __restrict__ rho,
                                                   float* __restrict__ sigma,
                                                   float* __restrict__ logsig_part) {
    __shared__ float red[256];
    const int tid = threadIdx.x;
    const int idx = blockIdx.x * 256 + tid;
    float r = rho[idx];
    float s = softplusf(r);
    sigma[idx] = s;
    red[tid] = __logf(s);
    __syncthreads();
    for (int st = 128; st >= 1; st >>= 1) {
        if (tid < st) red[tid] += red[tid + st];
        __syncthreads();
    }
    if (tid == 0) logsig_part[blockIdx.x] = red[0];
}

// ---------------------------------------------------------------------------
// Kernel 2: out[b,o] = sum_i x[b,i] * mu[o,i]   via V_WMMA_F32_16X16X4_F32
// Block = 128 threads = 4 waves. Each wave owns a 16x16 (MxN) tile.
// grid = (64/16, 1024/64) = (4, 16). K-loop unrolled 4 -> 4+ wmma per iter.
// A: lane l holds X[m0 + l%16][k + 2*(l/16) + j]  (j = 0,1)
// B: lane l holds mu[n0 + l%16][k + 2*(l/16) + j] (contiguous, row-major in i)
// D: VGPR r, lanes 0-15: (M=r, N=lane); lanes 16-31: (M=r+8, N=lane-16)
// ---------------------------------------------------------------------------
__global__ __launch_bounds__(128) void gemm_mu_kernel(const float* __restrict__ x,
                                                      const float* __restrict__ mu,
                                                      float* __restrict__ out) {
    const int lane = threadIdx.x & 31;
    const int wave = threadIdx.x >> 5;
    const int m0 = blockIdx.x * 16;             // batch-row tile
    const int n0 = blockIdx.y * 64 + wave * 16; // out-col tile
    const int l16 = lane & 15;
    const int kh  = (lane >> 4) << 1;           // 0 or 2

    const float* arow = x  + (m0 + l16) * 1024;
    const float* brow = mu + (n0 + l16) * 1024;

    v8f c = {};
    for (int k = 0; k < 1024; k += 16) {
#pragma unroll
        for (int kk = 0; kk < 16; kk += 4) {
            v2f a = *(const v2f*)(arow + k + kk + kh);
            v2f b = *(const v2f*)(brow + k + kk + kh);
            c = __builtin_amdgcn_wmma_f32_16x16x4_f32(
                    false, a, false, b, (short)0, c, false, false);
        }
    }

    const int mrow = m0 + ((lane >> 4) << 3);
#pragma unroll
    for (int r = 0; r < 8; ++r)
        out[(mrow + r) * 1024 + n0 + l16] = c[r];
}

// ---------------------------------------------------------------------------
// Kernel 3: stream eps_w (256 MB) once — HBM-bound (~11us floor @ 23.3 TB/s).
//   per element: t = sigma*eps; w = t + mu;
//     y2[b,o]    += t * x[b,i]          (stochastic part of einsum)
//     prior_acc  += mix_lp(w)
//     eps2_acc   += eps*eps             (posterior via eps^2/2 identity)
// eps_w is read once and exceeds the 192MB L2 -> non-temporal loads so the
// 64x-reused sigma/mu (4MB each) stay L2-resident. Per-lane b128 loads:
// 512B per wave per issue, fully coalesced, 1/4 the VMEM instruction count.
// Block = 256 threads (8 waves) owns one b and 16 o's (wave -> 2 o's, full i).
// ---------------------------------------------------------------------------
__global__ __launch_bounds__(256) void main_kernel(const float* __restrict__ x,
                                                   const float* __restrict__ mu,
                                                   const float* __restrict__ eps_w,
                                                   const float* __restrict__ sigma,
                                                   float* __restrict__ out,
                                                   float* __restrict__ prior_part,
                                                   float* __restrict__ eps2_part) {
    __shared__ float xs[1024];
    __shared__ float wred[16];
    const int tid = threadIdx.x;
    const int b  = blockIdx.x >> 6;
    const int o0 = (blockIdx.x & 63) << 4;

#pragma unroll
    for (int j = 0; j < 4; ++j) xs[tid + j * 256] = x[b * 1024 + tid + j * 256];
    __syncthreads();

    const int lane = tid & 31;
    const int wave = tid >> 5;
    float prior_acc = 0.f, eps2_acc = 0.f;

#pragma unroll
    for (int oo = 0; oo < 2; ++oo) {
        const int o = o0 + wave * 2 + oo;
        const float* er = eps_w + ((size_t)b * 1024 + (size_t)o) * 1024;
        const float* sr = sigma + o * 1024;
        const float* mr = mu + o * 1024;
        float y = 0.f;
        for (int it = 0; it < 8; ++it) {
            const int base = (it << 7) + (lane << 2);      // 16B-aligned
            v4f e4 = __builtin_nontemporal_load((const v4f*)(er + base));
            v4f s4 = *(const v4f*)(sr + base);
            v4f m4 = *(const v4f*)(mr + base);
            v4f x4 = *(const v4f*)(xs + base);
#pragma unroll
            for (int j = 0; j < 4; ++j) {
                float e = e4[j];
                float t = s4[j] * e;
                float w = t + m4[j];
                y = fmaf(t, x4[j], y);
                prior_acc += mix_lp(w);
                eps2_acc = fmaf(e, e, eps2_acc);
            }
        }
        // wave32 reduction of y
#pragma unroll
        for (int msk = 16; msk >= 1; msk >>= 1) y += __shfl_xor(y, msk, 32);
        if (lane == 0) out[b * 1024 + o] += y;   // unique owner -> plain RMW
    }

#pragma unroll
    for (int msk = 16; msk >= 1; msk >>= 1) {
        prior_acc += __shfl_xor(prior_acc, msk, 32);
        eps2_acc  += __shfl_xor(eps2_acc,  msk, 32);
    }
    if (lane == 0) { wred[wave] = prior_acc; wred[8 + wave] = eps2_acc; }
    __syncthreads();
    if (tid == 0) {
        float p = 0.f, e2 = 0.f;
#pragma unroll
        for (int w2 = 0; w2 < 8; ++w2) { p += wred[w2]; e2 += wred[8 + w2]; }
        prior_part[blockIdx.x] = p;
        eps2_part[blockIdx.x]  = e2;
    }
}

// ---------------------------------------------------------------------------
// Kernel 4: bias sample + its prior/posterior. 64K elements, 256 blocks x 256.
// ---------------------------------------------------------------------------
__global__ __launch_bounds__(256) void bias_kernel(const float* __restrict__ bias_mu,
                                                   const float* __restrict__ bias_rho,
                                                   const float* __restrict__ eps_b,
                                                   float* __restrict__ out,
                                                   float* __restrict__ bprior_part,
                                                   float* __restrict__ bpost_part) {
    __shared__ float redp[256];
    __shared__ float redq[256];
    const int tid = threadIdx.x;
    const int idx = blockIdx.x * 256 + tid;
    const int o = idx & 1023;
    float r = bias_rho[o];
    float s = softplusf(r);
    float e = __builtin_nontemporal_load(eps_b + idx);
    float bval = fmaf(s, e, bias_mu[o]);
    out[idx] += bval;
    redp[tid] = mix_lp(bval);
    redq[tid] = -LOG_SQRT_2PI - __logf(s) - 0.5f * e * e;
    __syncthreads();
    for (int st = 128; st >= 1; st >>= 1) {
        if (tid < st) { redp[tid] += redp[tid + st]; redq[tid] += redq[tid + st]; }
        __syncthreads();
    }
    if (tid == 0) { bprior_part[blockIdx.x] = redp[0]; bpost_part[blockIdx.x] = redq[0]; }
}

// ---------------------------------------------------------------------------
// Kernel 5: deterministic final reduction of all block partials.
// ---------------------------------------------------------------------------
__global__ __launch_bounds__(256) void reduce_kernel(const float* __restrict__ lg,
                                                     const float* __restrict__ pr,
                                                     const float* __restrict__ e2,
                                                     const float* __restrict__ bp,
                                                     const float* __restrict__ bq,
                                                     float* __restrict__ outs) {
    __shared__ float red[256];
    const int tid = threadIdx.x;
    float v[5] = {0.f, 0.f, 0.f, 0.f, 0.f};
    for (int i = tid; i < 4096; i += 256) { v[0] += lg[i]; v[1] += pr[i]; v[2] += e2[i]; }
    v[3] = bp[tid];
    v[4] = bq[tid];
    float res[5];
#pragma unroll
    for (int q = 0; q < 5; ++q) {
        red[tid] = v[q];
        __syncthreads();
        for (int st = 128; st >= 1; st >>= 1) {
            if (tid < st) red[tid] += red[tid + st];
            __syncthreads();
        }
        res[q] = red[0];
        __syncthreads();
    }
    if (tid == 0) {
        // log_prior = weight-prior + bias-prior
        outs[0] = res[1] + res[3];
        // log_posterior = -B*O*I*c - B*sum(log sigma_w) - 0.5*sum(eps_w^2) + bias part
        outs[1] = -64.0f * 1048576.0f * LOG_SQRT_2PI
                  - 64.0f * res[0] - 0.5f * res[2] + res[4];
    }
}

extern "C" void kernel_launch(void* const* d_in, const int* in_sizes, int n_in,
                              void* d_out, int out_size, void* d_ws, size_t ws_size,
                              hipStream_t stream) {
    const float* x    = (const float*)d_in[0];
    const float* wmu  = (const float*)d_in[1];
    const float* wrho = (const float*)d_in[2];
    const float* bmu  = (const float*)d_in[3];
    const float* brho = (const float*)d_in[4];
    const float* epsw = (const float*)d_in[5];
    const float* epsb = (const float*)d_in[6];
    float* out = (float*)d_out;

    // workspace layout (floats): sigma_w [1M] | logsig[4096] | prior[4096]
    //                            | eps2[4096] | bprior[256] | bpost[256]
    float* ws      = (float*)d_ws;
    float* sigma   = ws;
    float* logsigp = ws + (1 << 20);
    float* priorp  = logsigp + 4096;
    float* eps2p   = priorp + 4096;
    float* bpriorp = eps2p + 4096;
    float* bpostp  = bpriorp + 256;

    prep_kernel<<<4096, 256, 0, stream>>>(wrho, sigma, logsigp);
    gemm_mu_kernel<<<dim3(4, 16), 128, 0, stream>>>(x, wmu, out);
    main_kernel<<<4096, 256, 0, stream>>>(x, wmu, epsw, sigma, out, priorp, eps2p);
    bias_kernel<<<256, 256, 0, stream>>>(bmu, brho, epsb, out, bpriorp, bpostp);
    reduce_kernel<<<1, 256, 0, stream>>>(logsigp, priorp, eps2p, bpriorp, bpostp,
                                         out + 64 * 1024);
}